// HPUAttentionImpl_78323023610465
// MI455X (gfx1250) — compile-verified
//
#include <hip/hip_runtime.h>
#include <math.h>

// Problem constants (match reference)
#define B_    32
#define QH_   32
#define KVH_  8
#define G_    4
#define D_    128
#define BS_   128
#define NB_   512
#define CONSTV 10.0f
#define EPS_  1.1754943508222875e-38f
#define SCALE_ 0.08838834764831845f

// LDS row stride: 132 mod 64 = 4 -> 16 fragment rows hit 16 distinct banks,
// lane-halves (+2 float offset) hit the disjoint odd set -> conflict-free.
#define LSTR 132

typedef float v2f __attribute__((ext_vector_type(2)));
typedef float v8f __attribute__((ext_vector_type(8)));

// ---------------------------------------------------------------------------
// Pass 1: per (block n, kv-head) attention partials via fp32 WMMA 16x16x4.
// grid = NB*KVH workgroups, 128 threads (4 waves).
// ---------------------------------------------------------------------------
__global__ __launch_bounds__(128)
void pa_block_kernel(const float* __restrict__ query,        // [B,QH,D]
                     const float* __restrict__ key_cache,    // [NCB,BS,KVH,D]
                     const float* __restrict__ value_cache,  // [NCB,BS,KVH,D]
                     const float* __restrict__ block_mapping,// [NB,B]
                     const float* __restrict__ block_bias,   // [NB,BS]
                     const int*   __restrict__ block_list,   // [NB]
                     float* __restrict__ out_p,              // [NB,KVH,G,D]
                     float* __restrict__ bsum)               // [NB,KVH,G]
{
    __shared__ float k_s[64 * LSTR];   // 33,792 B : half of the K tile
    __shared__ float q_s[16 * LSTR];   //  8,448 B : Q rows 0..3, rest zero
    __shared__ float p_s[16 * LSTR];   //  8,448 B : exp-probs rows 0..3
    __shared__ float map_s[B_];

    const int tid  = threadIdx.x;
    const int lane = tid & 31;
    const int w    = tid >> 5;                 // wave id 0..3
    const int n    = blockIdx.x / KVH_;
    const int kvh  = blockIdx.x % KVH_;
    const int blk  = block_list[n];

    // zero padded LDS matrices; load mapping row
    for (int i = tid; i < 16 * LSTR; i += 128) { q_s[i] = 0.f; p_s[i] = 0.f; }
    if (tid < B_) map_s[tid] = block_mapping[(size_t)n * B_ + tid];
    __syncthreads();

    // q_s[g][d] = SCALE * sum_b map[b] * query[b, kvh*G+g, d]  (faithful einsum)
    for (int idx = tid; idx < G_ * D_; idx += 128) {
        const int g = idx >> 7, d = idx & (D_ - 1);
        const float* qp = query + (size_t)(kvh * G_ + g) * D_ + d;
        float acc = 0.f;
        for (int b = 0; b < B_; ++b) acc += map_s[b] * qp[(size_t)b * QH_ * D_];
        q_s[g * LSTR + d] = SCALE_ * acc;
    }
    __syncthreads();

    const int m16  = lane & 15;
    const int koff = (lane >> 4) << 1;               // 0 or 2
    const size_t rstride = (size_t)KVH_ * D_;        // row stride in caches
    const float* kbase = key_cache   + ((size_t)blk * BS_ * KVH_ + kvh) * D_;
    const float* vbase = value_cache + ((size_t)blk * BS_ * KVH_ + kvh) * D_;

    // ---------------- scores: two halves of 64 keys each -------------------
    for (int h = 0; h < 2; ++h) {
        // coalesced float4 stage of 64 key rows into LDS
        for (int it = 0; it < 16; ++it) {
            const int flat = it * 128 + tid;         // 2048 float4s
            const int s  = flat >> 5;                // local key row 0..63
            const int d4 = flat & 31;                // float4 within row
            const float4 v = *(const float4*)(kbase + (size_t)(h * 64 + s) * rstride + d4 * 4);
            *(float4*)(&k_s[s * LSTR + d4 * 4]) = v;
        }
        __syncthreads();

        // wave w -> key tile (h*4 + w): S[m=head][n=key] += Q[m][d] * K[key][d]
        v8f c = {0.f, 0.f, 0.f, 0.f, 0.f, 0.f, 0.f, 0.f};
        const int srow = w * 16 + m16;               // key row (local) = column n
        for (int d0 = 0; d0 < D_; d0 += 4) {
            v2f a, bf;
            a.x  = q_s[m16 * LSTR + d0 + koff];
            a.y  = q_s[m16 * LSTR + d0 + koff + 1];
            bf.x = k_s[srow * LSTR + d0 + koff];
            bf.y = k_s[srow * LSTR + d0 + koff + 1];
            c = __builtin_amdgcn_wmma_f32_16x16x4_f32(false, a, false, bf,
                                                      (short)0, c, false, false);
        }
        if (lane < 16) {   // rows 0..3 of D live in c[0..3], lanes 0..15 = key col
            const int key = (h * 4 + w) * 16 + lane;
            const float bias = block_bias[(size_t)n * BS_ + key];
            p_s[0 * LSTR + key] = __expf(c[0] + bias - CONSTV);
            p_s[1 * LSTR + key] = __expf(c[1] + bias - CONSTV);
            p_s[2 * LSTR + key] = __expf(c[2] + bias - CONSTV);
            p_s[3 * LSTR + key] = __expf(c[3] + bias - CONSTV);
        }
        __syncthreads();
    }

    // per-(g) block sums
    if (tid < G_) {
        float s = 0.f;
        for (int k = 0; k < BS_; ++k) s += p_s[tid * LSTR + k];
        bsum[((size_t)n * KVH_ + kvh) * G_ + tid] = s;
    }

    // ---------------- P @ V : wave w handles d-tiles {w, w+4} --------------
    for (int t = 0; t < 2; ++t) {
        const int dt0 = (t * 4 + w) * 16;
        v8f c = {0.f, 0.f, 0.f, 0.f, 0.f, 0.f, 0.f, 0.f};
        for (int k0 = 0; k0 < BS_; k0 += 4) {
            v2f a, bf;
            a.x = p_s[m16 * LSTR + k0 + koff];
            a.y = p_s[m16 * LSTR + k0 + koff + 1];
            // B[k=key][n=d]: half-wave reads 16 consecutive floats of one V row
            const float* vp = vbase + (size_t)(k0 + koff) * rstride + dt0 + m16;
            bf.x = vp[0];
            bf.y = vp[rstride];
            c = __builtin_amdgcn_wmma_f32_16x16x4_f32(false, a, false, bf,
                                                      (short)0, c, false, false);
        }
        if (lane < 16) {
            const size_t base = ((size_t)n * KVH_ + kvh) * G_ * D_ + dt0 + lane;
            out_p[base + 0 * D_] = c[0];
            out_p[base + 1 * D_] = c[1];
            out_p[base + 2 * D_] = c[2];
            out_p[base + 3 * D_] = c[3];
        }
    }
}

// ---------------------------------------------------------------------------
// Pass 2: seqsum[b,k,g] = sum_n map[n,b] * bsum[n,k,g]   (block2batch)
// ---------------------------------------------------------------------------
__global__ void seqsum_kernel(const float* __restrict__ block_mapping,
                              const float* __restrict__ bsum,
                              float* __restrict__ seqsum)
{
    const int idx = blockIdx.x * blockDim.x + threadIdx.x;
    if (idx >= B_ * KVH_ * G_) return;
    const int b = idx / (KVH_ * G_);
    const int kg = idx % (KVH_ * G_);
    float acc = 0.f;
    for (int n = 0; n < NB_; ++n)
        acc += block_mapping[(size_t)n * B_ + b] * bsum[(size_t)n * KVH_ * G_ + kg];
    seqsum[idx] = acc;
}

// ---------------------------------------------------------------------------
// Pass 3: inv_denom[n,k,g] = 1 / max(bsum, batch2block(seqsum) + EPS)
// ---------------------------------------------------------------------------
__global__ void invden_kernel(const float* __restrict__ block_mapping,
                              const float* __restrict__ bsum,
                              const float* __restrict__ seqsum,
                              float* __restrict__ invden)
{
    const int idx = blockIdx.x * blockDim.x + threadIdx.x;
    if (idx >= NB_ * KVH_ * G_) return;
    const int n = idx / (KVH_ * G_);
    const int kg = idx % (KVH_ * G_);
    float acc = 0.f;
    for (int b = 0; b < B_; ++b)
        acc += block_mapping[(size_t)n * B_ + b] * seqsum[(size_t)b * KVH_ * G_ + kg];
    const float denom = fmaxf(bsum[idx], acc + EPS_);
    invden[idx] = 1.0f / denom;
}

// ---------------------------------------------------------------------------
// Pass 4: out[b,h,d] = sum_n map[n,b] * out_p[n,h,d] * invden[n,h]
// ---------------------------------------------------------------------------
__global__ void finalize_kernel(const float* __restrict__ block_mapping,
                                const float* __restrict__ out_p,
                                const float* __restrict__ invden,
                                float* __restrict__ out)
{
    const int idx = blockIdx.x * blockDim.x + threadIdx.x;
    if (idx >= B_ * QH_ * D_) return;
    const int b = idx / (QH_ * D_);
    const int h = (idx / D_) % QH_;          // = kvh*G + g
    const int d = idx & (D_ - 1);
    float acc = 0.f;
    for (int n = 0; n < NB_; ++n) {
        const float m = block_mapping[(size_t)n * B_ + b];
        if (m != 0.f) {
            acc += m * out_p[((size_t)n * QH_ + h) * D_ + d]
                     * invden[(size_t)n * QH_ + h];
        }
    }
    out[idx] = acc;
}

// ---------------------------------------------------------------------------
extern "C" void kernel_launch(void* const* d_in, const int* in_sizes, int n_in,
                              void* d_out, int out_size, void* d_ws, size_t ws_size,
                              hipStream_t stream)
{
    const float* query         = (const float*)d_in[0];
    const float* key_cache     = (const float*)d_in[1];
    const float* value_cache   = (const float*)d_in[2];
    const float* block_mapping = (const float*)d_in[3];
    const float* block_bias    = (const float*)d_in[4];
    const int*   block_list    = (const int*)d_in[5];
    float* out = (float*)d_out;

    // workspace layout (floats)
    float* ws     = (float*)d_ws;
    float* out_p  = ws;                                   // NB*KVH*G*D = 2,097,152
    float* bsum   = out_p + (size_t)NB_ * KVH_ * G_ * D_; // NB*KVH*G   =    16,384
    float* seqsum = bsum + (size_t)NB_ * KVH_ * G_;       // B*KVH*G    =     1,024
    float* invden = seqsum + (size_t)B_ * KVH_ * G_;      // NB*KVH*G   =    16,384

    pa_block_kernel<<<NB_ * KVH_, 128, 0, stream>>>(
        query, key_cache, value_cache, block_mapping, block_bias, block_list,
        out_p, bsum);

    seqsum_kernel<<<(B_ * KVH_ * G_ + 255) / 256, 256, 0, stream>>>(
        block_mapping, bsum, seqsum);

    invden_kernel<<<(NB_ * KVH_ * G_ + 255) / 256, 256, 0, stream>>>(
        block_mapping, bsum, seqsum, invden);

    finalize_kernel<<<(B_ * QH_ * D_ + 255) / 256, 256, 0, stream>>>(
        block_mapping, out_p, invden, out);
}